// CombinedLoss_23201413333226
// MI455X (gfx1250) — compile-verified
//
#include <hip/hip_runtime.h>
#include <hip/hip_bf16.h>
#include <stdint.h>

typedef __attribute__((ext_vector_type(2))) float v2f;
typedef __attribute__((ext_vector_type(8))) float v8f;

#define B_ROWS 16384
#define C_CLS  1000
#define D_FEAT 512

#define CE_WAVES   8
#define CE_BLOCKS  (B_ROWS / CE_WAVES)                 // 2048
#define COL_WAVES  8
#define COL_BLOCKS (B_ROWS / (16 * COL_WAVES))         // 128
#define WS_COL_OFF 4096                                // float offset of collapse partials in ws

// ---------------------------------------------------------------------------
// Kernel 1: cross-entropy. One wave per row of outputs[16384,1000].
// Row (4000 B = 250 x b128) is staged into LDS with CDNA5 async loads,
// then two LDS passes: rowmax, sum(exp(x - max)); loss = max + log(S) - x[label].
// ---------------------------------------------------------------------------
__global__ __launch_bounds__(256) void ce_kernel(const float* __restrict__ outputs,
                                                 const int*   __restrict__ labels,
                                                 float*       __restrict__ partials)
{
    __shared__ float smem[CE_WAVES * C_CLS];   // 32000 B, 16B-aligned slots (4000 B stride)
    __shared__ float wloss[CE_WAVES];

    const int wave = threadIdx.x >> 5;
    const int lane = threadIdx.x & 31;
    const int row  = blockIdx.x * CE_WAVES + wave;

    // ---- async stage: global -> LDS, 250 b128 chunks per row ----
    const uint64_t gbase = (uint64_t)(uintptr_t)(outputs + (size_t)row * C_CLS);
    const uint32_t lbase = (uint32_t)(uintptr_t)(&smem[wave * C_CLS]); // low 32 bits = LDS offset
#pragma unroll
    for (int i = 0; i < 8; ++i) {
        const int chunk = lane + 32 * i;
        if (chunk < 250) {
            uint64_t ga = gbase + (uint64_t)chunk * 16u;
            uint32_t la = lbase + (uint32_t)chunk * 16u;
            asm volatile("global_load_async_to_lds_b128 %0, %1, off"
                         :: "v"(la), "v"(ga) : "memory");
        }
    }
    asm volatile("s_wait_asynccnt 0x0" ::: "memory");

    const float4* srow = (const float4*)(&smem[wave * C_CLS]);

    // ---- pass 1: row max ----
    float m = -INFINITY;
#pragma unroll
    for (int i = 0; i < 8; ++i) {
        const int idx = lane + 32 * i;
        if (idx < 250) {
            float4 v = srow[idx];
            m = fmaxf(m, fmaxf(fmaxf(v.x, v.y), fmaxf(v.z, v.w)));
        }
    }
#pragma unroll
    for (int off = 16; off; off >>= 1) m = fmaxf(m, __shfl_xor(m, off, 32));

    // ---- pass 2: sum of exp(x - max) ----
    float s = 0.0f;
#pragma unroll
    for (int i = 0; i < 8; ++i) {
        const int idx = lane + 32 * i;
        if (idx < 250) {
            float4 v = srow[idx];
            s += __expf(v.x - m) + __expf(v.y - m) + __expf(v.z - m) + __expf(v.w - m);
        }
    }
#pragma unroll
    for (int off = 16; off; off >>= 1) s += __shfl_xor(s, off, 32);

    if (lane == 0) {
        const int lab = labels[row];
        const float xl = smem[wave * C_CLS + lab];
        wloss[wave] = m + __logf(s) - xl;          // = -log_softmax[label]
    }
    __syncthreads();
    if (threadIdx.x == 0) {
        float t = 0.0f;
#pragma unroll
        for (int w = 0; w < CE_WAVES; ++w) t += wloss[w];
        partials[blockIdx.x] = t;
    }
}

// ---------------------------------------------------------------------------
// Kernel 2: neural-collapse margin. 16 rows per wave.
// diff = features[r] - means[label[r]] is fed as BOTH A and B of
// V_WMMA_F32_16X16X4_F32: for this shape the A (16x4) and B (4x16) lane/VGPR
// mappings coincide, so diag(D) accumulates sum_k diff[r][k]^2 per row.
// ---------------------------------------------------------------------------
__global__ __launch_bounds__(256) void collapse_kernel(const float* __restrict__ features,
                                                       const float* __restrict__ means,
                                                       const int*   __restrict__ labels,
                                                       float*       __restrict__ partials)
{
    __shared__ float wsum[COL_WAVES];
    const int wave = threadIdx.x >> 5;
    const int lane = threadIdx.x & 31;
    const int r    = (blockIdx.x * COL_WAVES + wave) * 16 + (lane & 15);
    const int coff = (lane >> 4) << 1;              // lanes 0-15: K+{0,1}; 16-31: K+{2,3}
    const int lab  = labels[r];

    const float* fp = features + (size_t)r   * D_FEAT + coff;
    const float* mp = means    + (size_t)lab * D_FEAT + coff;

    v8f c = {};
#pragma unroll 4
    for (int k = 0; k < D_FEAT; k += 4) {
        float2 f = *(const float2*)(fp + k);
        float2 g = *(const float2*)(mp + k);
        v2f a;
        a.x = f.x - g.x;
        a.y = f.y - g.y;
        c = __builtin_amdgcn_wmma_f32_16x16x4_f32(false, a, false, a,
                                                  (short)0, c, false, false);
    }

    // diag(c): row rr lives at VGPR (rr&7), lane rr (rr<8) or rr+16 (rr>=8).
    float acc = 0.0f;
#pragma unroll
    for (int rr = 0; rr < 16; ++rr) {
        float d2 = __shfl(c[rr & 7], (rr < 8) ? rr : (rr + 16), 32);
        acc += fmaxf(5.0f - sqrtf(d2), 0.0f);       // relu(eps - dist)
    }
    if (lane == 0) wsum[wave] = acc;
    __syncthreads();
    if (threadIdx.x == 0) {
        float t = 0.0f;
#pragma unroll
        for (int w = 0; w < COL_WAVES; ++w) t += wsum[w];
        partials[blockIdx.x] = t;
    }
}

// ---------------------------------------------------------------------------
// Kernel 3: deterministic final reduction -> scalar loss.
// out = (sum_ce + 0.1 * sum_col) / B
// ---------------------------------------------------------------------------
__global__ __launch_bounds__(256) void finalize_kernel(const float* __restrict__ ws,
                                                       float*       __restrict__ out)
{
    __shared__ float red[256];
    const int t = threadIdx.x;
    float ce = 0.0f;
#pragma unroll
    for (int i = 0; i < CE_BLOCKS / 256; ++i) ce += ws[t + i * 256];
    float col = (t < COL_BLOCKS) ? ws[WS_COL_OFF + t] : 0.0f;
    red[t] = ce + 0.1f * col;
    __syncthreads();
#pragma unroll
    for (int s = 128; s > 0; s >>= 1) {
        if (t < s) red[t] += red[t + s];
        __syncthreads();
    }
    if (t == 0) out[0] = red[0] * (1.0f / (float)B_ROWS);
}

extern "C" void kernel_launch(void* const* d_in, const int* in_sizes, int n_in,
                              void* d_out, int out_size, void* d_ws, size_t ws_size,
                              hipStream_t stream)
{
    const float* outputs  = (const float*)d_in[0];   // [16384,1000] f32
    const float* features = (const float*)d_in[1];   // [16384,512]  f32
    const float* means    = (const float*)d_in[2];   // [1000,512]   f32
    const int*   labels   = (const int*)  d_in[3];   // [16384]      i32
    float* ws  = (float*)d_ws;
    float* out = (float*)d_out;

    ce_kernel      <<<CE_BLOCKS,  256, 0, stream>>>(outputs, labels, ws);
    collapse_kernel<<<COL_BLOCKS, 256, 0, stream>>>(features, means, labels, ws + WS_COL_OFF);
    finalize_kernel<<<1,          256, 0, stream>>>(ws, out);
}